// GCN_2l_26740466385303
// MI455X (gfx1250) — compile-verified
//
#include <hip/hip_runtime.h>
#include <math.h>

// CDNA5 / gfx1250, wave32. fp32 WMMA path: V_WMMA_F32_16X16X4_F32.
typedef __attribute__((ext_vector_type(2))) float v2f;
typedef __attribute__((ext_vector_type(8))) float v8f;

#define DIM 128          // node / hidden feature dim (fixed by the model)
#define NCLS 40          // classes
#define NCLS_PAD 48      // padded to 3 x 16 column tiles

// ---------------------------------------------------------------- utilities
__global__ void k_zero(float* __restrict__ p, int n) {
  int i = blockIdx.x * blockDim.x + threadIdx.x;
  if (i < n) p[i] = 0.0f;
}

// deg[col[e]] += 1  (self-loop +1 folded into k_dinv)
__global__ void k_degree(const int* __restrict__ col, int E, float* __restrict__ deg) {
  int e = blockIdx.x * blockDim.x + threadIdx.x;
  if (e < E) atomicAdd(&deg[col[e]], 1.0f);
}

// in-place: deg -> 1/sqrt(deg + 1)   (deg+1 >= 1 always, no zero guard needed)
__global__ void k_dinv(float* __restrict__ deg, int n) {
  int i = blockIdx.x * blockDim.x + threadIdx.x;
  if (i < n) deg[i] = rsqrtf(deg[i] + 1.0f);
}

// ------------------------------------------------- WMMA GEMM:  D = A * W^T
// A: [nRows x DIM] row-major, W: [outDim x DIM] row-major, D: [nRows x outDim].
// One wave computes one 16x16 output tile, chaining 32 x V_WMMA_F32_16X16X4_F32.
//
// ISA layouts (CDNA5 ISA):
//  A 16x4 f32 : lane L<16 holds {A[m0+L][k], A[m0+L][k+1]},
//               lane L>=16 holds {A[m0+L-16][k+2], A[m0+L-16][k+3]}.
//  B 4x16 f32 : lane L<16 holds {B[k][j0+L], B[k+1][j0+L]},
//               lane L>=16 holds {B[k+2][j0+L-16], B[k+3][j0+L-16]}.
//  Since B[k][j] = W[j][k] (row-major W), both A and B loads are
//  "row (base + lane%16), columns k + 2*(lane/16) .. +1" -> identical pattern.
//  C/D 16x16  : VGPR v, lanes 0-15 -> row m0+v, lanes 16-31 -> row m0+v+8,
//               col j0 + lane%16.
// Requires nRows % 16 == 0 (N = 50000 = 3125 * 16).
__global__ void k_gemm_wmma(const float* __restrict__ A, const float* __restrict__ W,
                            float* __restrict__ D, int outDim, int rowTiles, int colTiles) {
  const int wave = threadIdx.x >> 5;
  const int lane = threadIdx.x & 31;
  const int t = blockIdx.x * (blockDim.x >> 5) + wave;
  if (t >= rowTiles * colTiles) return;
  const int m0 = (t / colTiles) << 4;
  const int j0 = (t % colTiles) << 4;

  const int lr = lane & 15;            // row within tile
  const int kh = (lane >> 4) << 1;     // 0 or 2: k sub-offset for this half-wave
  const float* __restrict__ arow = A + (size_t)(m0 + lr) * DIM;
  const float* __restrict__ brow = W + (size_t)(j0 + lr) * DIM;

  v8f c = {};
#pragma unroll
  for (int k = 0; k < DIM; k += 4) {
    v2f a = *(const v2f*)(arow + k + kh);   // 8B-aligned (k+kh even)
    v2f b = *(const v2f*)(brow + k + kh);
    c = __builtin_amdgcn_wmma_f32_16x16x4_f32(
        /*neg_a=*/false, a, /*neg_b=*/false, b,
        /*c_mod=*/(short)0, c, /*reuse_a=*/false, /*reuse_b=*/false);
  }

  const int rOff = (lane >> 4) << 3;   // 0 or 8
  float* __restrict__ drow = D + (size_t)(m0 + rOff) * outDim + j0 + lr;
#pragma unroll
  for (int v = 0; v < 8; ++v) drow[(size_t)v * outDim] = c[v];
}

// ----------------------------------------- self-loop init: agg = h * dinv^2
// One wave per node; lane covers 4 channels via float4 (32*4 = 128).
__global__ void k_selfinit(const float* __restrict__ h, const float* __restrict__ dinv,
                           float* __restrict__ agg, int n) {
  const int wave = threadIdx.x >> 5;
  const int lane = threadIdx.x & 31;
  const int i = blockIdx.x * (blockDim.x >> 5) + wave;
  if (i >= n) return;
  float s = dinv[i];
  s *= s;
  const float4* __restrict__ hp = (const float4*)(h + (size_t)i * DIM);
  float4* __restrict__ ap = (float4*)(agg + (size_t)i * DIM);
  float4 v = hp[lane];
  ap[lane] = make_float4(v.x * s, v.y * s, v.z * s, v.w * s);
}

// --------------------------- edge scatter: agg[col] += h[row] * dinv_r*dinv_c
// One wave per edge; lane covers 4 channels (float4 read, 4x atomic f32 add).
__global__ void k_scatter(const float* __restrict__ h, const float* __restrict__ dinv,
                          const int* __restrict__ row, const int* __restrict__ col,
                          float* __restrict__ agg, int E) {
  const int wave = threadIdx.x >> 5;
  const int lane = threadIdx.x & 31;
  const int e = blockIdx.x * (blockDim.x >> 5) + wave;
  if (e >= E) return;
  const int r = row[e];
  const int cn = col[e];
  const float s = dinv[r] * dinv[cn];
  const float4* __restrict__ hp = (const float4*)(h + (size_t)r * DIM);
  float4 v = hp[lane];
  float* __restrict__ ap = agg + (size_t)cn * DIM + lane * 4;
  atomicAdd(ap + 0, v.x * s);
  atomicAdd(ap + 1, v.y * s);
  atomicAdd(ap + 2, v.z * s);
  atomicAdd(ap + 3, v.w * s);
}

// ------------------------------------------------ h = relu(agg + b[channel])
__global__ void k_biasrelu(const float* __restrict__ agg, const float* __restrict__ b,
                           float* __restrict__ h, long long total) {
  long long i = (long long)blockIdx.x * blockDim.x + threadIdx.x;
  if (i < total) {
    float v = agg[i] + b[i & (DIM - 1)];
    h[i] = v > 0.0f ? v : 0.0f;
  }
}

// -------------------------------- pad Wf (40x128) -> Wfp (48x128, zero tail)
__global__ void k_padW(const float* __restrict__ Wf, float* __restrict__ Wfp) {
  int i = blockIdx.x * blockDim.x + threadIdx.x;
  if (i >= NCLS_PAD * DIM) return;
  int r = i / DIM;
  Wfp[i] = (r < NCLS) ? Wf[i] : 0.0f;
}

// ------------------ per-node log_softmax over 40 classes (+bias), wave32
// One wave per node: lane handles class `lane`, and class `lane+32` if lane<8.
__global__ void k_logsoftmax(const float* __restrict__ logits, const float* __restrict__ bf,
                             float* __restrict__ out, int n) {
  const int wave = threadIdx.x >> 5;
  const int lane = threadIdx.x & 31;
  const int i = blockIdx.x * (blockDim.x >> 5) + wave;
  if (i >= n) return;
  const float* __restrict__ lp = logits + (size_t)i * NCLS_PAD;
  const float NEG = -3.402823466e+38f;
  float x0 = lp[lane] + bf[lane];
  float x1 = (lane < 8) ? (lp[lane + 32] + bf[lane + 32]) : NEG;
  float m = fmaxf(x0, x1);
#pragma unroll
  for (int o = 16; o > 0; o >>= 1) m = fmaxf(m, __shfl_xor(m, o, 32));
  float s = expf(x0 - m) + ((lane < 8) ? expf(x1 - m) : 0.0f);
#pragma unroll
  for (int o = 16; o > 0; o >>= 1) s += __shfl_xor(s, o, 32);
  float lse = m + logf(s);
  float* __restrict__ op = out + (size_t)i * NCLS;
  op[lane] = x0 - lse;
  if (lane < 8) op[lane + 32] = x1 - lse;
}

// ---------------------------------------------------------------------------
extern "C" void kernel_launch(void* const* d_in, const int* in_sizes, int n_in,
                              void* d_out, int out_size, void* d_ws, size_t ws_size,
                              hipStream_t stream) {
  const float* x  = (const float*)d_in[0];
  const int*   ei = (const int*)d_in[1];   // [2, E] flat: row then col
  const float* W1 = (const float*)d_in[2];
  const float* b1 = (const float*)d_in[3];
  const float* W2 = (const float*)d_in[4];
  const float* b2 = (const float*)d_in[5];
  const float* Wf = (const float*)d_in[6];
  const float* bf = (const float*)d_in[7];

  const int n = in_sizes[0] / DIM;   // 50000 (multiple of 16)
  const int E = in_sizes[1] / 2;     // 600000
  const int* rowIdx = ei;
  const int* colIdx = ei + E;

  // ---- carve workspace (256 B aligned chunks)
  float* ws = (float*)d_ws;
  size_t off = 0;
  auto carve = [&](size_t cnt) -> float* {
    float* p = ws + off;
    off += (cnt + 63) & ~(size_t)63;
    return p;
  };
  float* dinv   = carve((size_t)n);             // degree, then 1/sqrt
  float* pre    = carve((size_t)n * DIM);       // GEMM output (pre-aggregation)
  float* agg    = carve((size_t)n * DIM);       // scatter accumulation
  float* h      = carve((size_t)n * DIM);       // post-relu activations
  float* logits = carve((size_t)n * NCLS_PAD);  // padded final logits
  float* Wfp    = carve((size_t)NCLS_PAD * DIM);
  (void)ws_size; (void)n_in; (void)out_size;

  const int rowT = n / 16;           // 3125
  const int colT = DIM / 16;         // 8
  const int tiles = rowT * colT;
  const int tilesF = rowT * (NCLS_PAD / 16);

  // ---- normalization: deg -> dinv
  k_zero<<<(n + 255) / 256, 256, 0, stream>>>(dinv, n);
  k_degree<<<(E + 255) / 256, 256, 0, stream>>>(colIdx, E, dinv);
  k_dinv<<<(n + 255) / 256, 256, 0, stream>>>(dinv, n);

  const long long tot = (long long)n * DIM;
  const int blkEl = (int)((tot + 255) / 256);

  // ---- layer 1
  k_gemm_wmma<<<(tiles + 3) / 4, 128, 0, stream>>>(x, W1, pre, DIM, rowT, colT);
  k_selfinit<<<(n + 7) / 8, 256, 0, stream>>>(pre, dinv, agg, n);
  k_scatter<<<(E + 7) / 8, 256, 0, stream>>>(pre, dinv, rowIdx, colIdx, agg, E);
  k_biasrelu<<<blkEl, 256, 0, stream>>>(agg, b1, h, tot);

  // ---- layer 2
  k_gemm_wmma<<<(tiles + 3) / 4, 128, 0, stream>>>(h, W2, pre, DIM, rowT, colT);
  k_selfinit<<<(n + 7) / 8, 256, 0, stream>>>(pre, dinv, agg, n);
  k_scatter<<<(E + 7) / 8, 256, 0, stream>>>(pre, dinv, rowIdx, colIdx, agg, E);
  k_biasrelu<<<blkEl, 256, 0, stream>>>(agg, b2, h, tot);

  // ---- classifier + log_softmax
  k_padW<<<(NCLS_PAD * DIM + 255) / 256, 256, 0, stream>>>(Wf, Wfp);
  k_gemm_wmma<<<(tilesF + 3) / 4, 128, 0, stream>>>(h, Wfp, logits, NCLS_PAD, rowT, NCLS_PAD / 16);
  k_logsoftmax<<<(n + 3) / 4, 128, 0, stream>>>(logits, bf, (float*)d_out, n);
}